// VoxelScatter_1425929142949
// MI455X (gfx1250) — compile-verified
//
#include <hip/hip_runtime.h>

// PillarScatter for MI455X (gfx1250).
//
// Reference: canvas = zeros([B*L, C]); canvas[b*L+lin] = vf[n]; -> [B, C, L].
// So out[b, c, lin] = (map[b*L+lin] >= 0) ? vf[map[b*L+lin], c] : 0
// with lin = z*NX*NY + y*NX + x.
//
// Strategy: build an inverse index map (pillar id per canvas cell) in scratch,
// then do a fully-coalesced streaming gather-write of the 512 MiB output.
// All output stores are 128B-coalesced nontemporal wave32 stores; all feature
// reads are read-once 256B-contiguous b128 gathers with NT hints.
// Roofline: 512 MiB NT writes + ~205 MB gather reads + ~16 MiB map traffic
// ~= 0.73 GiB -> ~32 us at 23.3 TB/s. Zero FLOPs, so no WMMA applies.

typedef __attribute__((ext_vector_type(4))) float v4f;  // native vector for NT builtins

constexpr int kNX = 512;
constexpr int kNY = 512;
constexpr int kNZ = 1;
constexpr int kC  = 64;
constexpr int kL  = kNX * kNY * kNZ;   // 262144 = 2^18
constexpr int kLshift = 18;            // log2(kL)

// ---------------------------------------------------------------------------
// Pass 1: map[i] = -1 for all B*L cells (int4-vectorized, 8 MiB).
// ---------------------------------------------------------------------------
__global__ void __launch_bounds__(256)
vs_init_map(int4* __restrict__ map4, int n4) {
    int i = blockIdx.x * 256 + threadIdx.x;
    if (i < n4) {
        map4[i] = make_int4(-1, -1, -1, -1);
    }
}

// ---------------------------------------------------------------------------
// Pass 2: map[b*L + lin] = pillar index n.  One thread per pillar.
// coords row = (b, z, y, x), int32, 16B-aligned -> single b128 load.
// Scattered 4B writes, but only 3.2 MB total and the 8 MiB map lives in L2.
// Indices are unique per batch (reference guarantees it), so plain stores.
// ---------------------------------------------------------------------------
__global__ void __launch_bounds__(256)
vs_scatter_idx(const int4* __restrict__ coords, int* __restrict__ map, int n) {
    int i = blockIdx.x * 256 + threadIdx.x;
    if (i < n) {
        int4 c = coords[i];  // (b, z, y, x)
        int lin = c.y * (kNX * kNY) + c.z * kNX + c.w;
        map[(c.x << kLshift) + lin] = i;
    }
}

// ---------------------------------------------------------------------------
// Pass 3 (dominant): one thread per (b, lin) output column.
//  - map load: coalesced, read-once -> nontemporal
//  - feature row: 16x global_load_b128 NT, 256 contiguous bytes per lane,
//    each vf row is read exactly once across the whole grid
//  - 64 channel stores: per-channel the wave writes 128 contiguous bytes ->
//    perfectly coalesced global_store_b32 with NT hint (streaming output,
//    never re-read; keep it out of the 192 MB L2's retained set)
// No cross-lane data exchange needed: a lane owns all C channels of its lin.
// ---------------------------------------------------------------------------
__global__ void __launch_bounds__(256)
vs_gather_write(const float* __restrict__ vf,
                const int* __restrict__ map,
                float* __restrict__ out) {
    const int t   = blockIdx.x * 256 + threadIdx.x;  // t in [0, B*L)
    const int b   = t >> kLshift;
    const int lin = t & (kL - 1);

    const int idx = __builtin_nontemporal_load(map + t);

    v4f r[kC / 4];
    if (idx >= 0) {
        const v4f* row = reinterpret_cast<const v4f*>(vf) + idx * (kC / 4);
#pragma unroll
        for (int j = 0; j < kC / 4; ++j) {
            r[j] = __builtin_nontemporal_load(row + j);
        }
    } else {
        const v4f zero = {0.0f, 0.0f, 0.0f, 0.0f};
#pragma unroll
        for (int j = 0; j < kC / 4; ++j) {
            r[j] = zero;
        }
    }

    float* ob = out + (((size_t)b * kC) << kLshift) + lin;
    const float* rf = reinterpret_cast<const float*>(r);
#pragma unroll
    for (int c = 0; c < kC; ++c) {
        __builtin_nontemporal_store(rf[c], ob + ((size_t)c << kLshift));
    }
}

// ---------------------------------------------------------------------------
// Launch: init map -> scatter indices -> streaming gather-write.
// d_ws must hold B*L ints (8 MiB for B=8); it is fully re-initialized every
// call, so the launch is deterministic and capture-safe.
// ---------------------------------------------------------------------------
extern "C" void kernel_launch(void* const* d_in, const int* in_sizes, int n_in,
                              void* d_out, int out_size, void* d_ws, size_t ws_size,
                              hipStream_t stream) {
    const float* vf     = (const float*)d_in[0];  // [N, 64] f32
    const int*   coords = (const int*)d_in[1];    // [N, 4] i32

    const int n     = in_sizes[1] / 4;            // number of pillars
    const int B     = out_size / (kC * kL);       // batch size from output shape
    const int total = B * kL;                     // canvas cells (2^21 for B=8)

    int* map = (int*)d_ws;                        // needs total*4 bytes (8 MiB)

    const int n4 = total / 4;
    vs_init_map<<<(n4 + 255) / 256, 256, 0, stream>>>((int4*)map, n4);
    vs_scatter_idx<<<(n + 255) / 256, 256, 0, stream>>>(
        (const int4*)coords, map, n);
    vs_gather_write<<<total / 256, 256, 0, stream>>>(
        vf, map, (float*)d_out);
}